// ROIPooling_40656160424512
// MI455X (gfx1250) — compile-verified
//
#include <hip/hip_runtime.h>
#include <stdint.h>

// ROI adaptive max pool, S=7, shapes fixed by the reference harness:
// feature_map [B=2, C=256, W=64, H=64] f32, rois [B=2, R=256, 4] f32,
// out [B, R, C, 7, 7] f32.
#define S_   7
#define B_   2
#define C_   256
#define W_   64
#define H_   64
#define R_   256

__global__ __launch_bounds__(256) void roi_pool_s7(
    const float* __restrict__ fm,
    const float* __restrict__ rois,
    float* __restrict__ out)
{
    // LDS staging tile in exact output order for this ROI: [c][i][j]
    __shared__ __align__(16) float sm[C_ * S_ * S_];   // 12544 floats = 50176 B

    const int br  = blockIdx.x;          // b*R + r
    const int b   = br / R_;
    const int c   = threadIdx.x;         // one channel per thread (C_ == blockDim)

    // ROI box (uniform across the block); float values are exact small ints.
    const float4 roi = *(const float4*)(rois + 4 * br);
    const int x1 = (int)roi.x;
    const int y1 = (int)roi.y;
    const int x2 = (int)roi.z;
    const int y2 = (int)roi.w;
    const int nx = x2 - x1 + 1;
    const int ny = y2 - y1 + 1;

    // Adaptive bin bounds along y (torch semantics: [lo+floor(j*n/S), lo+ceil((j+1)*n/S)) )
    int sy[S_], ey[S_];
#pragma unroll
    for (int j = 0; j < S_; ++j) {
        sy[j] = y1 + (j * ny) / S_;
        ey[j] = y1 + ((j + 1) * ny + (S_ - 1)) / S_;
    }

    const float* fmb = fm + (size_t)(b * C_ + c) * (W_ * H_);
    float* smc = sm + c * (S_ * S_);

#pragma unroll
    for (int i = 0; i < S_; ++i) {
        const int xs = x1 + (i * nx) / S_;
        const int xe = x1 + ((i + 1) * nx + (S_ - 1)) / S_;
        float acc[S_];
#pragma unroll
        for (int j = 0; j < S_; ++j) acc[j] = -__FLT_MAX__;   // == finfo(f32).min
        for (int x = xs; x < xe; ++x) {
            const float* rp = fmb + x * H_;                    // contiguous in y
#pragma unroll
            for (int j = 0; j < S_; ++j) {
                for (int y = sy[j]; y < ey[j]; ++y)
                    acc[j] = fmaxf(acc[j], rp[y]);
            }
        }
#pragma unroll
        for (int j = 0; j < S_; ++j) smc[i * S_ + j] = acc[j]; // ds_store, bank-conflict-free
    }

    __syncthreads();

    // ---- CDNA5 async LDS -> global DMA of the whole contiguous output tile ----
    // 12544 floats: 12 x (256 lanes x b128) + 1 x (256 lanes x b32).
    const uint32_t lds0 = (uint32_t)(uintptr_t)(&sm[0]);       // flat low 32b == LDS offset
    const uint64_t g0   = (uint64_t)(uintptr_t)(out + (size_t)br * (C_ * S_ * S_));
    const int tid = threadIdx.x;
    constexpr int NB128 = (C_ * S_ * S_) / 4 / 256;            // 12

#pragma unroll
    for (int it = 0; it < NB128; ++it) {
        const uint32_t off = (uint32_t)((it * 256 + tid) * 16);
        const uint64_t ga  = g0 + off;
        const uint32_t la  = lds0 + off;
        asm volatile("global_store_async_from_lds_b128 %0, %1, off"
                     :: "v"(ga), "v"(la) : "memory");
    }
    {
        const uint32_t off = (uint32_t)(NB128 * 256 * 16 + tid * 4);
        const uint64_t ga  = g0 + off;
        const uint32_t la  = lds0 + off;
        asm volatile("global_store_async_from_lds_b32 %0, %1, off"
                     :: "v"(ga), "v"(la) : "memory");
    }
    // LDS is deallocated at wave end: drain the async DMA before s_endpgm.
    asm volatile("s_wait_asynccnt 0" ::: "memory");
}

extern "C" void kernel_launch(void* const* d_in, const int* in_sizes, int n_in,
                              void* d_out, int out_size, void* d_ws, size_t ws_size,
                              hipStream_t stream) {
    (void)in_sizes; (void)n_in; (void)d_ws; (void)ws_size; (void)out_size;
    const float* fm   = (const float*)d_in[0];
    const float* rois = (const float*)d_in[1];
    float*       out  = (float*)d_out;
    dim3 grid(B_ * R_);   // one block per ROI
    dim3 block(C_);       // one thread per channel (8 wave32s)
    roi_pool_s7<<<grid, block, 0, stream>>>(fm, rois, out);
}